// ProbeCorrectionModule_24824910971216
// MI455X (gfx1250) — compile-verified
//
#include <hip/hip_runtime.h>

typedef __attribute__((ext_vector_type(2))) float v2f;
typedef __attribute__((ext_vector_type(8))) float v8f;

#define B_DIM 8
#define N_DIM 65536
#define P_DIM 64
#define TILES_PER_B (N_DIM / 16)
#define WAVES_PER_BLOCK 8

// XOR-butterfly lane shuffle within 32-lane groups (xor masks 1,2,4,8 stay
// inside each 16-lane half): ds_swizzle offset = {xor[14:10], or[9:5], and[4:0]}
#define SWZ_XOR(v, m)                                                          \
  __builtin_bit_cast(float, __builtin_amdgcn_ds_swizzle(                       \
                                __builtin_bit_cast(int, (v)),                  \
                                (0x1f | ((m) << 10))))

#define RED16(v)                                                               \
  do {                                                                         \
    (v) += SWZ_XOR((v), 1);                                                    \
    (v) += SWZ_XOR((v), 2);                                                    \
    (v) += SWZ_XOR((v), 4);                                                    \
    (v) += SWZ_XOR((v), 8);                                                    \
  } while (0)

// Accumulate 4 WMMA tiles; c2 = -gamma*log2(e) is pre-folded into bf, so the
// WMMA result is already the exp2 argument. MASKED path multiplies by the
// per-row validity mask; the unmasked path is taken (wave-uniformly) whenever
// the whole 16-row tile is in range.
template <bool MASKED>
__device__ __forceinline__ void accumulate_tiles(
    const v2f a, const v2f* __restrict__ bf, const float* __restrict__ rx,
    const float* __restrict__ ry, const float* __restrict__ maskf,
    float* __restrict__ sw, float* __restrict__ sx, float* __restrict__ sy) {
#pragma unroll
  for (int t = 0; t < 4; ++t) {
    v8f c = {};
    v8f e = __builtin_amdgcn_wmma_f32_16x16x4_f32(
        /*neg_a=*/false, a, /*neg_b=*/false, bf[t],
        /*c_mod=*/(short)0, c, /*reuse_a=*/false, /*reuse_b=*/false);
#pragma unroll
    for (int r = 0; r < 8; ++r) {
      float w = __builtin_amdgcn_exp2f(e[r]); // e[r] = c2 * d2
      if (MASKED) w *= maskf[r];
      sw[r] += w;
      sx[r] += w * rx[t];
      sy[r] += w * ry[t];
    }
  }
}

__global__ __launch_bounds__(WAVES_PER_BLOCK * 32)
void probe_correction_wmma_kernel(const float* __restrict__ y_base,
                                  const float* __restrict__ branch_real,
                                  const float* __restrict__ trunk,
                                  const float* __restrict__ probe_true,
                                  const float* __restrict__ gamma_ptr,
                                  const long long* __restrict__ lengths,
                                  float* __restrict__ out) {
  const int lane = threadIdx.x & 31;
  const int wave = threadIdx.x >> 5;
  const int tile = blockIdx.x * WAVES_PER_BLOCK + wave; // one 16-row tile/wave
  const int b    = tile / TILES_PER_B;
  const int n0   = (tile - b * TILES_PER_B) * 16;

  const bool hi = (lane >= 16);
  const int  lm = lane & 15;

  const float gamma = gamma_ptr[0];
  const float c2    = -gamma * 1.4426950408889634f; // -gamma * log2(e)
  const int   len   = (int)lengths[b];

  // ---- A fragment: row M = lm; lo lanes (K0,K1)=(tx,ty), hi (K2,K3)=(|t|^2,1)
  const float* trow = trunk + ((size_t)b * N_DIM + (size_t)(n0 + lm)) * 3;
  const float tx = trow[0];
  const float ty = trow[1];
  v2f a;
  a.x = hi ? (tx * tx + ty * ty) : tx;
  a.y = hi ? 1.0f : ty;

  // ---- B fragments (4 probe tiles of 16), scaled by c2, + residuals ----
  v2f   bf[4];
  float rx[4], ry[4];
#pragma unroll
  for (int t = 0; t < 4; ++t) {
    const int    p  = lm + t * 16;
    const float* br = branch_real + (size_t)b * (P_DIM * 5) + (size_t)p * 5;
    const float  px = br[0];
    const float  py = br[1];
    // Column p of B, pre-scaled: c2 * (-2px, -2py, 1, |p|^2)
    bf[t].x = hi ? c2 : (c2 * -2.0f * px);
    bf[t].y = hi ? (c2 * (px * px + py * py)) : (c2 * -2.0f * py);

    const float2 pt = *(const float2*)(probe_true + ((size_t)b * P_DIM + p) * 2);
    const float2 yb = *(const float2*)(y_base + ((size_t)b * N_DIM + p) * 2);
    rx[t] = pt.x - yb.x;
    ry[t] = pt.y - yb.y;
  }

  // validity mask per D-row register r (row M = r + 8*hi)
  float maskf[8];
#pragma unroll
  for (int r = 0; r < 8; ++r) {
    const int n = n0 + r + (hi ? 8 : 0);
    maskf[r]    = (n < len) ? 1.0f : 0.0f;
  }

  float sw[8], sx[8], sy[8];
#pragma unroll
  for (int r = 0; r < 8; ++r) { sw[r] = 0.0f; sx[r] = 0.0f; sy[r] = 0.0f; }

  // Wave-uniform branch: fully-valid tiles skip the mask multiply entirely.
  if (n0 + 16 <= len) {
    accumulate_tiles<false>(a, bf, rx, ry, maskf, sw, sx, sy);
  } else {
    accumulate_tiles<true>(a, bf, rx, ry, maskf, sw, sx, sy);
  }

  // ---- reduce over the probe dimension (16-lane butterfly per half) ----
#pragma unroll
  for (int r = 0; r < 8; ++r) {
    RED16(sw[r]);
    RED16(sx[r]);
    RED16(sy[r]);
  }

  // ---- normalize + write: lane lm==r writes row n0 + r + 8*hi ----
#pragma unroll
  for (int r = 0; r < 8; ++r) {
    if (lm == r) {
      const int    n   = n0 + r + (hi ? 8 : 0);
      const float  inv = 1.0f / fmaxf(sw[r], 1e-8f);
      const size_t idx = ((size_t)b * N_DIM + (size_t)n) * 2;
      const float2 yb  = *(const float2*)(y_base + idx);
      float2 o;
      o.x = yb.x + sx[r] * inv;
      o.y = yb.y + sy[r] * inv;
      *(float2*)(out + idx) = o;
    }
  }
}

extern "C" void kernel_launch(void* const* d_in, const int* in_sizes, int n_in,
                              void* d_out, int out_size, void* d_ws,
                              size_t ws_size, hipStream_t stream) {
  (void)in_sizes; (void)n_in; (void)out_size; (void)d_ws; (void)ws_size;
  const float*     y_base      = (const float*)d_in[0];
  const float*     branch_real = (const float*)d_in[1];
  const float*     trunk       = (const float*)d_in[2];
  const float*     probe_true  = (const float*)d_in[3];
  const float*     gamma       = (const float*)d_in[4];
  const long long* lengths     = (const long long*)d_in[5];
  float*           out         = (float*)d_out;

  const int total_tiles = B_DIM * TILES_PER_B;          // 32768 waves
  dim3 block(WAVES_PER_BLOCK * 32);                     // 8 waves / block
  dim3 grid(total_tiles / WAVES_PER_BLOCK);             // 4096 blocks
  probe_correction_wmma_kernel<<<grid, block, 0, stream>>>(
      y_base, branch_real, trunk, probe_true, gamma, lengths, out);
}